// NeuralNetwork_24730421690845
// MI455X (gfx1250) — compile-verified
//
#include <hip/hip_runtime.h>

typedef __attribute__((ext_vector_type(2))) float v2f;
typedef __attribute__((ext_vector_type(8))) float v8f;

#define N        2048
#define NLAYERS  15
#define WAVES    16                // waves per block (wave32) -> 512 threads
#define BLOCK    (WAVES * 32)
#define KCHUNK   (N / WAVES)       // 128 floats of K per wave

// One layer: vout = act(W @ vin + b), W row-major N x N.
// Each block owns 16 output rows; 16 waves split the K dimension.
// Matvec runs through V_WMMA_F32_16X16X4_F32 with B's 16 columns all equal
// to the activation chunk, so every column of D carries the same 16 partial
// sums (redundant FLOPs are free; the kernel is HBM-streaming-bound on W).
__global__ __launch_bounds__(BLOCK) void layer_kernel(
    const float* __restrict__ W,
    const float* __restrict__ b,
    const float* __restrict__ vin,
    float* __restrict__ vout,
    int apply_silu)
{
    __shared__ float s_v[N];                // staged input vector (8 KB)
    __shared__ float s_part[WAVES][16];     // per-wave partial sums

    const int tid  = threadIdx.x;
    const int lane = tid & 31;
    const int wave = tid >> 5;
    const int row0 = blockIdx.x * 16;

    // Stage activation vector into LDS, coalesced across the whole block.
    for (int i = tid; i < N; i += BLOCK) s_v[i] = vin[i];
    __syncthreads();

    const int half = lane >> 4;    // 0: lanes 0-15, 1: lanes 16-31
    const int m    = lane & 15;    // A-fragment row within the 16-row tile
    const int koff = 2 * half;     // A vgpr0 holds K=0 (half 0) / K=2 (half 1)
    const int k0   = wave * KCHUNK;

    const float* __restrict__ wrow = W + (size_t)(row0 + m) * N;

    // 4 independent accumulators to break the WMMA D->C serial chain.
    v8f acc0 = {}, acc1 = {}, acc2 = {}, acc3 = {};

    #pragma unroll
    for (int c = k0; c < k0 + KCHUNK; c += 16) {
        // ---- 4 K-chunks of 4 ----
        v2f a0, a1, a2, a3, b0, b1, b2, b3;
        a0.x = wrow[c +  0 + koff];     a0.y = wrow[c +  1 + koff];
        a1.x = wrow[c +  4 + koff];     a1.y = wrow[c +  5 + koff];
        a2.x = wrow[c +  8 + koff];     a2.y = wrow[c +  9 + koff];
        a3.x = wrow[c + 12 + koff];     a3.y = wrow[c + 13 + koff];
        b0.x = s_v[c +  0 + koff];      b0.y = s_v[c +  1 + koff];
        b1.x = s_v[c +  4 + koff];      b1.y = s_v[c +  5 + koff];
        b2.x = s_v[c +  8 + koff];      b2.y = s_v[c +  9 + koff];
        b3.x = s_v[c + 12 + koff];      b3.y = s_v[c + 13 + koff];

        acc0 = __builtin_amdgcn_wmma_f32_16x16x4_f32(false, a0, false, b0,
                                                     (short)0, acc0, false, false);
        acc1 = __builtin_amdgcn_wmma_f32_16x16x4_f32(false, a1, false, b1,
                                                     (short)0, acc1, false, false);
        acc2 = __builtin_amdgcn_wmma_f32_16x16x4_f32(false, a2, false, b2,
                                                     (short)0, acc2, false, false);
        acc3 = __builtin_amdgcn_wmma_f32_16x16x4_f32(false, a3, false, b3,
                                                     (short)0, acc3, false, false);
    }

    v8f acc = (acc0 + acc1) + (acc2 + acc3);

    // C/D 16x16 f32 layout: lanes 0-15 vgpr g -> row g; lanes 16-31 -> row 8+g.
    // All 16 columns identical, so lanes 0 and 16 carry the full 16-row result.
    if (lane == 0 || lane == 16) {
        #pragma unroll
        for (int g = 0; g < 8; ++g)
            s_part[wave][half * 8 + g] = acc[g];
    }
    __syncthreads();

    // Reduce across waves, add bias, apply activation.
    if (tid < 16) {
        float sum = b[row0 + tid];
        #pragma unroll
        for (int w = 0; w < WAVES; ++w) sum += s_part[w][tid];
        if (apply_silu) sum = sum / (1.0f + __expf(-sum));   // silu
        vout[row0 + tid] = sum;
    }
}

extern "C" void kernel_launch(void* const* d_in, const int* in_sizes, int n_in,
                              void* d_out, int out_size, void* d_ws, size_t ws_size,
                              hipStream_t stream) {
    const float* x = (const float*)d_in[0];          // (N,)
    const float* W = (const float*)d_in[1];          // (15, N, N)
    const float* b = (const float*)d_in[2];          // (15, N)
    float* out = (float*)d_out;                      // (N,)

    float* ws0 = (float*)d_ws;                       // ping-pong activations
    float* ws1 = ws0 + N;

    const float* vin = x;
    for (int l = 0; l < NLAYERS; ++l) {
        float* vout = (l == NLAYERS - 1) ? out : ((l & 1) ? ws1 : ws0);
        const int silu = (l < NLAYERS - 1) ? 1 : 0;
        layer_kernel<<<N / 16, BLOCK, 0, stream>>>(
            W + (size_t)l * N * N, b + (size_t)l * N, vin, vout, silu);
        vin = vout;
    }
}